// ButterflyMatrix_81389630259270
// MI455X (gfx1250) — compile-verified
//
#include <hip/hip_runtime.h>

#define BLOCK 256
#define COLS 4096
#define DEPTH 4                    // async pipeline stages (power of 2)

// Native clang vector so __builtin_nontemporal_store accepts it.
typedef __attribute__((ext_vector_type(4))) float f4;

// LDS byte offset of a __shared__ object (generic -> addrspace(3) -> int).
__device__ __forceinline__ unsigned lds_offset(const void* p) {
    return (unsigned)(unsigned long long)
        (__attribute__((address_space(3))) const void*)p;
}

// gfx1250 async global->LDS 128-bit copy, non-temporal (streaming) hint.
__device__ __forceinline__ void async_copy_b128(const f4* gptr, unsigned lds_byte_off) {
    asm volatile("global_load_async_to_lds_b128 %0, %1, off th:TH_LOAD_NT"
                 :
                 : "v"(lds_byte_off), "v"((unsigned long long)gptr)
                 : "memory");
}

__device__ __forceinline__ void wait_async_le(int ahead) {
    // Wait until at most `ahead` async copies are still outstanding.
    if (ahead >= DEPTH - 1) { asm volatile("s_wait_asynccnt 0x3" ::: "memory"); }
    else if (ahead == 2)    { asm volatile("s_wait_asynccnt 0x2" ::: "memory"); }
    else if (ahead == 1)    { asm volatile("s_wait_asynccnt 0x1" ::: "memory"); }
    else                    { asm volatile("s_wait_asynccnt 0x0" ::: "memory"); }
}

__global__ __launch_bounds__(BLOCK) void ButterflyMatrix_kernel(
    const float* __restrict__ x,
    const float* __restrict__ tw,
    float* __restrict__ y,
    int iters,          // rows handled by this block (uniform across blocks)
    int rowStride4)     // float4 stride between successive rows of this block
{
    __shared__ __align__(16) float tile[DEPTH][BLOCK * 4];

    const int tid   = threadIdx.x;
    const int f4col = blockIdx.x * BLOCK + tid;      // this thread's float4 column slot

    // Two 2x2 twiddle blocks for columns [4*f4col .. 4*f4col+3]; register-cached
    // for the whole row loop (twiddle layout [k][2][2] is float4-granular).
    const f4 tA = ((const f4*)tw)[2 * f4col + 0];    // pair k   : {T00,T01,T10,T11}
    const f4 tB = ((const f4*)tw)[2 * f4col + 1];    // pair k+1

    const int f4_per_row = COLS / 4;
    const f4* __restrict__ xp = (const f4*)x + (size_t)blockIdx.y * f4_per_row + f4col;
    f4* __restrict__       yp = (f4*)y       + (size_t)blockIdx.y * f4_per_row + f4col;

    const unsigned ldsBase   = lds_offset(&tile[0][0]) + (unsigned)tid * 16u;
    const unsigned bufStride = (unsigned)BLOCK * 16u;

    // ---- prologue: fill the async pipeline (up to DEPTH loads in flight) ----
    const f4* xIssue = xp;
    #pragma unroll
    for (int d = 0; d < DEPTH; ++d) {
        if (d < iters) {
            async_copy_b128(xIssue, ldsBase + (unsigned)d * bufStride);
            xIssue += rowStride4;
        }
    }

    // ---- steady state: wait oldest, consume, re-issue into same buffer ----
    for (int i = 0; i < iters; ++i) {
        wait_async_le(iters - i - 1);                // oldest copy (buffer i%DEPTH) done

        const unsigned bufOff = (unsigned)(i & (DEPTH - 1)) * bufStride;
        const f4 v = *(const f4*)&tile[0][(bufOff >> 2) + (unsigned)tid * 4u];

        f4 o;
        o.x = tA.x * v.x + tA.y * v.y;   // pair k   row 0
        o.y = tA.z * v.x + tA.w * v.y;   // pair k   row 1
        o.z = tB.x * v.z + tB.y * v.w;   // pair k+1 row 0
        o.w = tB.z * v.z + tB.w * v.w;   // pair k+1 row 1

        __builtin_nontemporal_store(o, yp);          // global_store_b128 th:TH_STORE_NT
        yp += rowStride4;

        if (i + DEPTH < iters) {                     // refill this buffer slot
            async_copy_b128(xIssue, ldsBase + bufOff);
            xIssue += rowStride4;
        }
    }
}

extern "C" void kernel_launch(void* const* d_in, const int* in_sizes, int n_in,
                              void* d_out, int out_size, void* d_ws, size_t ws_size,
                              hipStream_t stream) {
    const float* x  = (const float*)d_in[0];
    const float* tw = (const float*)d_in[1];
    float*       y  = (float*)d_out;

    const int total = in_sizes[0];          // 4*2048*4096
    const int rows  = total / COLS;         // 8192 flattened rows
    const int f4_per_row = COLS / 4;        // 1024

    // Row-block count: largest power of two <= 512 dividing rows, so every
    // block runs a uniform iteration count (needed for immediate waits).
    int step = 512;
    while (step > 1 && (rows % step)) step >>= 1;
    const int iters = rows / step;          // 16 for the reference shapes

    dim3 grid(f4_per_row / BLOCK, step);    // (4, 512) = 2048 blocks, 16384 waves
    dim3 block(BLOCK);

    ButterflyMatrix_kernel<<<grid, block, 0, stream>>>(
        x, tw, y, iters, step * f4_per_row);
}